// Simple3DGS_21552145891475
// MI455X (gfx1250) — compile-verified
//
#include <hip/hip_runtime.h>

typedef float v2f __attribute__((ext_vector_type(2)));
typedef float v8f __attribute__((ext_vector_type(8)));

#define N_PTS   512
#define IMG     256
#define TILE    16
#define FX_     300.0f
#define FY_     300.0f
#define CX_     128.0f
#define CY_     128.0f
#define EPS2D_  0.3f
#define NEAR_   0.01f
#define C0_     0.28209479177387814f
#define PSTRIDE 12          // floats per gaussian record: mx,my,cA,cB,cC,op,r,g,b,pad,pad,pad

// ---------------------------------------------------------------------------
// Phase 1: per-gaussian projection -> conic / color / opacity / depth key
// ---------------------------------------------------------------------------
__global__ __launch_bounds__(256) void gs_preprocess(
    const float* __restrict__ cam,    const float* __restrict__ means,
    const float* __restrict__ quats,  const float* __restrict__ scales,
    const float* __restrict__ opacs,  const float* __restrict__ sh0,
    float* __restrict__ P, float* __restrict__ keys, int n)
{
  int i = blockIdx.x * blockDim.x + threadIdx.x;
  if (i >= n) return;

  // camtoworld = [A | t]; viewmat = diag(1,-1,-1) * [A^-1 | -A^-1 t]
  float a00=cam[0],a01=cam[1],a02=cam[2],t0=cam[3];
  float a10=cam[4],a11=cam[5],a12=cam[6],t1=cam[7];
  float a20=cam[8],a21=cam[9],a22=cam[10],t2=cam[11];
  float det3 = a00*(a11*a22-a12*a21) - a01*(a10*a22-a12*a20) + a02*(a10*a21-a11*a20);
  float rd = 1.0f/det3;
  float i00 =  (a11*a22-a12*a21)*rd, i01 = -(a01*a22-a02*a21)*rd, i02 =  (a01*a12-a02*a11)*rd;
  float i10 = -(a10*a22-a12*a20)*rd, i11 =  (a00*a22-a02*a20)*rd, i12 = -(a00*a12-a02*a10)*rd;
  float i20 =  (a10*a21-a11*a20)*rd, i21 = -(a00*a21-a01*a20)*rd, i22 =  (a00*a11-a01*a10)*rd;
  // rows 1,2 sign-flipped
  float R00= i00, R01= i01, R02= i02;
  float R10=-i10, R11=-i11, R12=-i12;
  float R20=-i20, R21=-i21, R22=-i22;
  float tv0 = -(i00*t0 + i01*t1 + i02*t2);
  float tv1 =  (i10*t0 + i11*t1 + i12*t2);
  float tv2 =  (i20*t0 + i21*t1 + i22*t2);

  float mx_ = means[i*3+0], my_ = means[i*3+1], mz_ = means[i*3+2];
  float x = R00*mx_ + R01*my_ + R02*mz_ + tv0;
  float y = R10*mx_ + R11*my_ + R12*mz_ + tv1;
  float z = R20*mx_ + R21*my_ + R22*mz_ + tv2;
  bool valid = z > NEAR_;
  float zs = valid ? z : 1.0f;
  float rz = 1.0f / zs;

  float qw = quats[i*4+0], qx = quats[i*4+1], qy = quats[i*4+2], qz = quats[i*4+3];
  float qn = rsqrtf(qw*qw+qx*qx+qy*qy+qz*qz);
  qw*=qn; qx*=qn; qy*=qn; qz*=qn;
  float Q00 = 1.f-2.f*(qy*qy+qz*qz), Q01 = 2.f*(qx*qy-qw*qz), Q02 = 2.f*(qx*qz+qw*qy);
  float Q10 = 2.f*(qx*qy+qw*qz), Q11 = 1.f-2.f*(qx*qx+qz*qz), Q12 = 2.f*(qy*qz-qw*qx);
  float Q20 = 2.f*(qx*qz-qw*qy), Q21 = 2.f*(qy*qz+qw*qx), Q22 = 1.f-2.f*(qx*qx+qy*qy);
  float e0 = __expf(scales[i*3+0]), e1 = __expf(scales[i*3+1]), e2 = __expf(scales[i*3+2]);
  float M00=Q00*e0, M01=Q01*e1, M02=Q02*e2;
  float M10=Q10*e0, M11=Q11*e1, M12=Q12*e2;
  float M20=Q20*e0, M21=Q21*e1, M22=Q22*e2;
  float S00=M00*M00+M01*M01+M02*M02;
  float S01=M00*M10+M01*M11+M02*M12;
  float S02=M00*M20+M01*M21+M02*M22;
  float S11=M10*M10+M11*M11+M12*M12;
  float S12=M10*M20+M11*M21+M12*M22;
  float S22=M20*M20+M21*M21+M22*M22;
  float T00=R00*S00+R01*S01+R02*S02, T01=R00*S01+R01*S11+R02*S12, T02=R00*S02+R01*S12+R02*S22;
  float T10=R10*S00+R11*S01+R12*S02, T11=R10*S01+R11*S11+R12*S12, T12=R10*S02+R11*S12+R12*S22;
  float T20=R20*S00+R21*S01+R22*S02, T21=R20*S01+R21*S11+R22*S12, T22=R20*S02+R21*S12+R22*S22;
  float V00=T00*R00+T01*R01+T02*R02;
  float V01=T00*R10+T01*R11+T02*R12;
  float V02=T00*R20+T01*R21+T02*R22;
  float V11=T10*R10+T11*R11+T12*R12;
  float V12=T10*R20+T11*R21+T12*R22;
  float V22=T20*R20+T21*R21+T22*R22;

  float limx = 1.3f*(0.5f*(float)IMG/FX_);
  float limy = 1.3f*(0.5f*(float)IMG/FY_);
  float tx = zs * fminf(fmaxf(x*rz,-limx),limx);
  float ty = zs * fminf(fmaxf(y*rz,-limy),limy);
  float J00 = FX_*rz, J02 = -FX_*tx*rz*rz;
  float J11 = FY_*rz, J12 = -FY_*ty*rz*rz;
  float A0 = J00*V00 + J02*V02;
  float A1 = J00*V01 + J02*V12;
  float A2 = J00*V02 + J02*V22;
  float B1 = J11*V11 + J12*V12;
  float B2 = J11*V12 + J12*V22;
  float c2_00 = A0*J00 + A2*J02;
  float c2_01 = A1*J11 + A2*J12;
  float c2_11 = B1*J11 + B2*J12;

  float aa = c2_00 + EPS2D_;
  float bb = c2_01;
  float cc = c2_11 + EPS2D_;
  float dd = aa*cc - bb*bb;
  valid = valid && (dd > 1e-12f);
  float ds = valid ? dd : 1.0f;
  float cA = cc/ds, cB = -bb/ds, cC = aa/ds;
  float px = FX_*x*rz + CX_;
  float py = FY_*y*rz + CY_;
  float op = valid ? (1.0f/(1.0f+__expf(-opacs[i]))) : 0.0f;  // valid mask folded in
  float cr = fmaxf(C0_*sh0[i*3+0]+0.5f, 0.0f);
  float cg = fmaxf(C0_*sh0[i*3+1]+0.5f, 0.0f);
  float cb = fmaxf(C0_*sh0[i*3+2]+0.5f, 0.0f);
  float* o = P + i*PSTRIDE;
  o[0]=px; o[1]=py; o[2]=cA; o[3]=cB; o[4]=cC; o[5]=op; o[6]=cr; o[7]=cg; o[8]=cb;
  o[9]=0.f; o[10]=0.f; o[11]=0.f;
  keys[i] = valid ? z : __builtin_inff();
}

// ---------------------------------------------------------------------------
// Phase 2: bitonic sort by (depth, idx) in LDS, then gather sorted records
// ---------------------------------------------------------------------------
__global__ __launch_bounds__(N_PTS) void gs_sort_gather(
    const float* __restrict__ keys, const float* __restrict__ P,
    float* __restrict__ SP, int n)
{
  __shared__ float sk[N_PTS];
  __shared__ int   si[N_PTS];
  int t = threadIdx.x;
  sk[t] = (t < n) ? keys[t] : __builtin_inff();
  si[t] = t;
  __syncthreads();
  for (int size = 2; size <= N_PTS; size <<= 1) {
    for (int stride = size >> 1; stride > 0; stride >>= 1) {
      int j = t ^ stride;
      if (j > t) {
        float k1 = sk[t], k2 = sk[j];
        int   i1 = si[t], i2 = si[j];
        bool ascending = ((t & size) == 0);
        bool gt = (k1 > k2) || (k1 == k2 && i1 > i2);   // strict total order
        if (ascending ? gt : !gt) {
          sk[t]=k2; sk[j]=k1; si[t]=i2; si[j]=i1;
        }
      }
      __syncthreads();
    }
  }
  int src = si[t];
  #pragma unroll
  for (int c = 0; c < PSTRIDE; ++c)
    SP[t*PSTRIDE + c] = (src < n) ? P[src*PSTRIDE + c] : 0.0f;
}

// ---------------------------------------------------------------------------
// Phase 3: tile render. Async-stage sorted table to LDS, sequential
// compositing, WMMA (f32 16x16x4) color accumulation.
// ---------------------------------------------------------------------------
__global__ __launch_bounds__(256) void gs_render(
    const float* __restrict__ SP, float* __restrict__ out)
{
  __shared__ float Pl[N_PTS * PSTRIDE];   // 24576 B sorted gaussian table
  __shared__ float Wl[256 * 17];          // 17408 B w-chunk / output staging (pad 17: conflict-free)
  const int t = threadIdx.x;

  // ---- CDNA5 async global->LDS staging (ASYNCcnt path) ----
  {
    unsigned ldsBase = (unsigned)(unsigned long long)(void*)&Pl[0];
    #pragma unroll
    for (int c = 0; c < (N_PTS*PSTRIDE*4)/(16*256); ++c) {   // 6 x b128 per thread
      unsigned byteOff = (unsigned)((c*256 + t) * 16);
      unsigned lds = ldsBase + byteOff;
      asm volatile("global_load_async_to_lds_b128 %0, %1, %2"
                   :: "v"(lds), "v"(byteOff), "s"(SP) : "memory");
    }
    asm volatile("s_wait_asynccnt 0" ::: "memory");
  }
  __syncthreads();

  const float px = (float)(blockIdx.x * TILE + (t & 15)) + 0.5f;
  const float py = (float)(blockIdx.y * TILE + (t >> 4)) + 0.5f;

  const int lane = t & 31;
  const int wave = t >> 5;
  const int m    = lane & 15;   // A: M row index, B: N column index
  const int kp   = lane >> 4;   // K-pair select (lanes 16-31 hold K=2,3)

  float T = 1.0f;
  v8f acc0 = {0.f,0.f,0.f,0.f,0.f,0.f,0.f,0.f};   // pixels wave*32 + 0..15
  v8f acc1 = {0.f,0.f,0.f,0.f,0.f,0.f,0.f,0.f};   // pixels wave*32 + 16..31

  for (int c = 0; c < N_PTS/16; ++c) {
    float wv[16];
    #pragma unroll
    for (int k = 0; k < 16; ++k) {
      const float* q = &Pl[(c*16 + k) * PSTRIDE];
      float dx = px - q[0];
      float dy = py - q[1];
      float sg = 0.5f*(q[2]*dx*dx + q[4]*dy*dy) + q[3]*dx*dy;
      float al = fminf(0.999f, q[5] * __expf(-sg));
      al = (sg >= 0.0f && al >= (1.0f/255.0f)) ? al : 0.0f;
      wv[k] = T * al;            // w = T_excl * alpha
      T *= (1.0f - al);
    }
    #pragma unroll
    for (int k = 0; k < 16; ++k) Wl[t*17 + k] = wv[k];
    __syncthreads();

    #pragma unroll
    for (int s4 = 0; s4 < 4; ++s4) {
      int kb = 4*s4 + 2*kp;
      int g0 = c*16 + kb;
      v2f a0, a1, b;
      a0.x = Wl[(wave*32 +      m)*17 + kb];
      a0.y = Wl[(wave*32 +      m)*17 + kb + 1];
      a1.x = Wl[(wave*32 + 16 + m)*17 + kb];
      a1.y = Wl[(wave*32 + 16 + m)*17 + kb + 1];
      b.x  = (m < 3) ? Pl[ g0   *PSTRIDE + 6 + m] : 0.0f;
      b.y  = (m < 3) ? Pl[(g0+1)*PSTRIDE + 6 + m] : 0.0f;
      acc0 = __builtin_amdgcn_wmma_f32_16x16x4_f32(false, a0, false, b, (short)0, acc0, false, false);
      acc1 = __builtin_amdgcn_wmma_f32_16x16x4_f32(false, a1, false, b, (short)0, acc1, false, false);
    }
    __syncthreads();
  }

  // extract D (M=pixel-in-group, N=channel) through LDS
  #pragma unroll
  for (int v = 0; v < 8; ++v) {
    int Mr = kp*8 + v;
    Wl[(wave*32 +      Mr)*17 + m] = acc0[v];
    Wl[(wave*32 + 16 + Mr)*17 + m] = acc1[v];
  }
  __syncthreads();

  float rr = Wl[t*17 + 0] + T;   // + T_final * BG, BG = 1.0
  float gg = Wl[t*17 + 1] + T;
  float bb = Wl[t*17 + 2] + T;
  int yy = blockIdx.y * TILE + (t >> 4);
  int xx = blockIdx.x * TILE + (t & 15);
  int pix = yy * IMG + xx;
  out[pix*3 + 0] = rr;
  out[pix*3 + 1] = gg;
  out[pix*3 + 2] = bb;
  out[IMG*IMG*3 + pix] = 1.0f - T;
}

// ---------------------------------------------------------------------------
extern "C" void kernel_launch(void* const* d_in, const int* in_sizes, int n_in,
                              void* d_out, int out_size, void* d_ws, size_t ws_size,
                              hipStream_t stream) {
  const float* cam    = (const float*)d_in[0];
  const float* means  = (const float*)d_in[1];
  const float* quats  = (const float*)d_in[2];
  const float* scales = (const float*)d_in[3];
  const float* opacs  = (const float*)d_in[4];
  const float* sh0    = (const float*)d_in[5];
  int n = in_sizes[4];
  if (n > N_PTS) n = N_PTS;

  float* ws   = (float*)d_ws;
  float* P    = ws;                         // 512*12 floats
  float* SP   = ws + N_PTS*PSTRIDE;         // 512*12 floats (sorted)
  float* keys = ws + 2*N_PTS*PSTRIDE;       // 512 floats

  gs_preprocess<<<(N_PTS + 255)/256, 256, 0, stream>>>(cam, means, quats, scales, opacs, sh0, P, keys, n);
  gs_sort_gather<<<1, N_PTS, 0, stream>>>(keys, P, SP, n);
  gs_render<<<dim3(IMG/TILE, IMG/TILE), 256, 0, stream>>>(SP, (float*)d_out);
}